// PtModule_76166950027602
// MI455X (gfx1250) — compile-verified
//
#include <hip/hip_runtime.h>
#include <hip/hip_bf16.h>

// y = (2x + 1)^2, elementwise over 8192x8192 fp32.
// Pure streaming kernel: 512 MB HBM traffic, ~128 MFLOP -> bandwidth bound.
// Strategy: float4 (b128) nontemporal loads/stores, grid-stride loop,
// FMA + square in registers. No LDS, no WMMA (no matrix op exists here).

typedef float __attribute__((ext_vector_type(4))) floatx4;

__global__ __launch_bounds__(256) void elemwise_sq_kernel(const float* __restrict__ x,
                                                          float* __restrict__ y,
                                                          long long n) {
    const long long nvec = n >> 2;  // number of float4 chunks
    const long long stride = (long long)gridDim.x * blockDim.x;
    long long i = (long long)blockIdx.x * blockDim.x + threadIdx.x;

    const floatx4* __restrict__ xv = (const floatx4*)x;
    floatx4* __restrict__ yv = (floatx4*)y;

    for (; i < nvec; i += stride) {
        // NT load: stream-once data, don't pollute L2 (512 MB stream vs 192 MB L2)
        floatx4 v = __builtin_nontemporal_load(&xv[i]);
        floatx4 r;
        #pragma unroll
        for (int k = 0; k < 4; ++k) {
            float t = __builtin_fmaf(v[k], 2.0f, 1.0f);  // v_fma_f32
            r[k] = t * t;                                 // v_mul_f32
        }
        __builtin_nontemporal_store(r, &yv[i]);
    }

    // Scalar tail (n not multiple of 4); 8192^2 has no tail, but stay correct.
    long long tail = nvec << 2;
    long long ti = tail + (long long)blockIdx.x * blockDim.x + threadIdx.x;
    if (ti < n) {
        float t = __builtin_fmaf(x[ti], 2.0f, 1.0f);
        y[ti] = t * t;
    }
}

extern "C" void kernel_launch(void* const* d_in, const int* in_sizes, int n_in,
                              void* d_out, int out_size, void* d_ws, size_t ws_size,
                              hipStream_t stream) {
    (void)n_in; (void)d_ws; (void)ws_size; (void)out_size;

    const float* x = (const float*)d_in[0];
    float* y = (float*)d_out;
    long long n = (long long)in_sizes[0];  // 8192*8192 = 67,108,864

    const int block = 256;                  // 8 wave32 waves per block
    long long nvec = (n + 3) >> 2;          // 16M float4 chunks
    long long want = (nvec + block - 1) / block;
    int grid = (int)(want < 16384 ? want : 16384);  // grid-stride; ~4 iters/thread
    if (grid < 1) grid = 1;

    elemwise_sq_kernel<<<grid, block, 0, stream>>>(x, y, n);
}